// AttnDecoderRNN_10565619548593
// MI455X (gfx1250) — compile-verified
//
#include <hip/hip_runtime.h>
#include <math.h>

static constexpr int kV = 32000;
static constexpr int kH = 256;
static constexpr int kS = 2048;
static constexpr int kB = 128;

typedef float v2f __attribute__((ext_vector_type(2)));
typedef float v8f __attribute__((ext_vector_type(8)));

__device__ __forceinline__ void wait_async_zero() {
#if __has_builtin(__builtin_amdgcn_s_wait_asynccnt)
    __builtin_amdgcn_s_wait_asynccnt(0);
#else
    asm volatile("s_wait_asynccnt 0" ::: "memory");
#endif
}

// ---------------------------------------------------------------------------
// K1: u[b,k] = sum_h h[b,h]*We[k,h];  bias[b] = sum_k h[b,k]*(h@Wh.T + attn_b)[k]
// attn_W is (H, 2H) row-major: Wh = cols [0,H), We = cols [H,2H)
// ---------------------------------------------------------------------------
__global__ void k1_attn_pre(const float* __restrict__ hidden,
                            const float* __restrict__ attn_W,
                            const float* __restrict__ attn_b,
                            float* __restrict__ u,
                            float* __restrict__ bias)
{
    const int b = blockIdx.x;
    const int k = threadIdx.x;           // 256 threads = one output k each
    __shared__ __align__(16) float hv[kH];
    __shared__ float red[256];
    hv[k] = hidden[b * kH + k];
    __syncthreads();

    const float4* hv4 = (const float4*)hv;
    const float4* wh4 = (const float4*)(attn_W + (size_t)k * (2 * kH));
    const float4* we4 = (const float4*)(attn_W + (size_t)k * (2 * kH) + kH);
    float ph = 0.f, uk = 0.f;
    #pragma unroll 4
    for (int i = 0; i < kH / 4; ++i) {
        float4 h4 = hv4[i];
        float4 a = wh4[i];
        float4 c = we4[i];
        ph += h4.x * a.x + h4.y * a.y + h4.z * a.z + h4.w * a.w;
        uk += h4.x * c.x + h4.y * c.y + h4.z * c.z + h4.w * c.w;
    }
    ph += attn_b[k];
    u[b * kH + k] = uk;
    red[k] = hv[k] * ph;
    __syncthreads();
    for (int off = 128; off > 0; off >>= 1) {
        if (k < off) red[k] += red[k + off];
        __syncthreads();
    }
    if (k == 0) bias[b] = red[0];
}

// ---------------------------------------------------------------------------
// K2 (fused, flash-style): single pass over enc computes scores AND an
// online-softmax-weighted context partial per S-split. Halves enc HBM traffic.
// Grid (B, 8): block owns 256 s-values; wave owns 32; lane owns an 8-float
// h-slice (as two float4 => global_load_b128, fully coalesced).
// Butterfly reduce leaves identical score in all lanes, so (m,d,ctx) stay
// lane-consistent with zero extra communication.
// ---------------------------------------------------------------------------
__global__ void k2_attn_fused(const float* __restrict__ enc,
                              const float* __restrict__ u,
                              const float* __restrict__ bias,
                              float* __restrict__ scores,
                              float* __restrict__ splitM,
                              float* __restrict__ splitD,
                              float* __restrict__ splitCtx)
{
    const int b    = blockIdx.x;
    const int j    = blockIdx.y;              // S-split index (8 splits x 256)
    const int lane = threadIdx.x & 31;
    const int wave = threadIdx.x >> 5;
    const int s0   = j * 256 + wave * 32;

    const float4* u4 = (const float4*)(u + b * kH);
    const float4 ua = u4[lane];
    const float4 ub = u4[32 + lane];
    const float bb = bias[b];

    float m = -INFINITY, d = 0.f;
    float4 ca = {0.f, 0.f, 0.f, 0.f};
    float4 cb = {0.f, 0.f, 0.f, 0.f};
    float myscore = 0.f;

    for (int si = 0; si < 32; ++si) {
        const int s = s0 + si;
        const float* ep = enc + ((size_t)s * kB + b) * kH;
        __builtin_prefetch(ep + (size_t)kB * kH, 0, 0);   // next s-row
        const float4* e4 = (const float4*)ep;
        const float4 ea = e4[lane];
        const float4 eb = e4[32 + lane];
        float dot = ea.x * ua.x + ea.y * ua.y + ea.z * ua.z + ea.w * ua.w
                  + eb.x * ub.x + eb.y * ub.y + eb.z * ub.z + eb.w * ub.w;
        #pragma unroll
        for (int off = 16; off > 0; off >>= 1) dot += __shfl_xor(dot, off, 32);
        const float score = dot + bb;
        if (lane == si) myscore = score;

        const float newm  = fmaxf(m, score);
        const float scale = __expf(m - newm);     // 0 on first iteration
        const float p     = __expf(score - newm);
        d = d * scale + p;
        ca.x = ca.x * scale + p * ea.x;  ca.y = ca.y * scale + p * ea.y;
        ca.z = ca.z * scale + p * ea.z;  ca.w = ca.w * scale + p * ea.w;
        cb.x = cb.x * scale + p * eb.x;  cb.y = cb.y * scale + p * eb.y;
        cb.z = cb.z * scale + p * eb.z;  cb.w = cb.w * scale + p * eb.w;
        m = newm;
    }
    scores[(size_t)b * kS + s0 + lane] = myscore;  // coalesced, 32 per wave

    // combine the 8 waves of this block
    __shared__ float wm[8], wd[8];
    __shared__ __align__(16) float wctx[8][kH];
    float4* wrow = (float4*)wctx[wave];
    wrow[lane]      = ca;
    wrow[32 + lane] = cb;
    if (lane == 0) { wm[wave] = m; wd[wave] = d; }
    __syncthreads();

    const int t = threadIdx.x;                    // t = h index
    float M = wm[0];
    #pragma unroll
    for (int w = 1; w < 8; ++w) M = fmaxf(M, wm[w]);
    float cs = 0.f, dsum = 0.f;
    #pragma unroll
    for (int w = 0; w < 8; ++w) {
        const float e = __expf(wm[w] - M);
        cs   += e * wctx[w][t];
        dsum += e * wd[w];
    }
    splitCtx[(size_t)(b * 8 + j) * kH + t] = cs;
    if (t == 0) { splitM[b * 8 + j] = M; splitD[b * 8 + j] = dsum; }
}

// ---------------------------------------------------------------------------
// K3: combine the 8 S-splits -> context[b,:]; emit normalized attn_weights
// from the stored scores (1 MB read instead of a 256 MB second enc pass).
// ---------------------------------------------------------------------------
__global__ void k3_combine(const float* __restrict__ scores,
                           const float* __restrict__ splitM,
                           const float* __restrict__ splitD,
                           const float* __restrict__ splitCtx,
                           float* __restrict__ attn_out,
                           float* __restrict__ ctx)
{
    const int b = blockIdx.x;
    const int t = threadIdx.x;                    // t = h index

    float M = splitM[b * 8];
    #pragma unroll
    for (int jj = 1; jj < 8; ++jj) M = fmaxf(M, splitM[b * 8 + jj]);
    float D = 0.f, c = 0.f;
    #pragma unroll
    for (int jj = 0; jj < 8; ++jj) {
        const float e = __expf(splitM[b * 8 + jj] - M);
        D += e * splitD[b * 8 + jj];
        c += e * splitCtx[(size_t)(b * 8 + jj) * kH + t];
    }
    const float inv = 1.f / D;
    ctx[b * kH + t] = c * inv;

    #pragma unroll
    for (int jj = 0; jj < 8; ++jj) {
        const int s = t + 256 * jj;
        attn_out[(size_t)b * kS + s] = __expf(scores[(size_t)b * kS + s] - M) * inv;
    }
}

// ---------------------------------------------------------------------------
// K4: GRU cell.  x = [embedding[ids[b]], context[b]] (512); gates -> h_new
// ---------------------------------------------------------------------------
__global__ void k4_gru(const int* __restrict__ ids,
                       const float* __restrict__ hidden,
                       const float* __restrict__ emb,
                       const float* __restrict__ ctx,
                       const float* __restrict__ W_ih,
                       const float* __restrict__ W_hh,
                       const float* __restrict__ b_ih,
                       const float* __restrict__ b_hh,
                       float* __restrict__ h_new)
{
    const int b = blockIdx.x;
    const int k = threadIdx.x;           // 256 threads = one output unit each
    __shared__ __align__(16) float x[2 * kH];
    __shared__ __align__(16) float hv[kH];
    const int id = ids[b];
    x[k]      = emb[(size_t)id * kH + k];
    x[kH + k] = ctx[b * kH + k];
    hv[k]     = hidden[b * kH + k];
    __syncthreads();

    float gi[3], gh[3];
    #pragma unroll
    for (int g = 0; g < 3; ++g) {
        const float4* wr = (const float4*)(W_ih + (size_t)(g * kH + k) * (2 * kH));
        const float4* xv = (const float4*)x;
        float acc = 0.f;
        for (int i = 0; i < (2 * kH) / 4; ++i) {
            float4 a = wr[i], c = xv[i];
            acc += a.x * c.x + a.y * c.y + a.z * c.z + a.w * c.w;
        }
        gi[g] = acc + b_ih[g * kH + k];

        const float4* wr2 = (const float4*)(W_hh + (size_t)(g * kH + k) * kH);
        const float4* hv4 = (const float4*)hv;
        float acc2 = 0.f;
        for (int i = 0; i < kH / 4; ++i) {
            float4 a = wr2[i], c = hv4[i];
            acc2 += a.x * c.x + a.y * c.y + a.z * c.z + a.w * c.w;
        }
        gh[g] = acc2 + b_hh[g * kH + k];
    }
    const float r = 1.f / (1.f + __expf(-(gi[0] + gh[0])));
    const float z = 1.f / (1.f + __expf(-(gi[1] + gh[1])));
    const float n = tanhf(gi[2] + r * gh[2]);
    h_new[b * kH + k] = (1.f - z) * n + z * hv[k];
}

// ---------------------------------------------------------------------------
// K5: logits = [h_new, ctx] (128x512) @ out_W (512x32000) + out_b
// WMMA f32 16x16x4. Wave owns a 16-wide N strip, accumulates all 8 M-tiles.
// A operand staged in LDS as padded [b][k] (stride 68 -> conflict-free
// ds_load_b64 fragments), filled by GLOBAL_LOAD_ASYNC_TO_LDS_B128 (ASYNCcnt).
// ---------------------------------------------------------------------------
__global__ void k5_logits_wmma(const float* __restrict__ h_new,
                               const float* __restrict__ ctx,
                               const float* __restrict__ out_W,
                               const float* __restrict__ out_b,
                               float* __restrict__ logits)
{
    constexpr int LDW = 68;                      // padded K-stride (words)
    __shared__ __align__(16) float catL[kB * LDW];   // [b][kk], kk in chunk
    const int t    = threadIdx.x;
    const int lane = t & 31;
    const int wave = t >> 5;
    const int half = lane >> 4;                  // K half: {0,1} vs {2,3}
    const int nl   = lane & 15;                  // N within tile / M within A
    const int n0   = blockIdx.x * 128 + wave * 16 + nl;

    v8f acc[8];
    #pragma unroll
    for (int mm = 0; mm < 8; ++mm)
        #pragma unroll
        for (int r = 0; r < 8; ++r) acc[mm][r] = 0.f;

    // staging assignment: thread -> (row bb, 32-float half-row)
    const int bb    = t >> 1;
    const int fbase = (t & 1) * 32;
    const unsigned ldsRow =
        (unsigned)(size_t)(void*)&catL[bb * LDW + fbase];

    for (int kc = 0; kc < 8; ++kc) {             // 8 chunks x 64 = K=512
        __syncthreads();                         // prior chunk fully consumed
        const float* src = (kc < 4 ? h_new + bb * kH + kc * 64
                                   : ctx   + bb * kH + (kc - 4) * 64) + fbase;
        #pragma unroll
        for (int i = 0; i < 8; ++i) {            // 8 x b128 = 32 floats
            const unsigned long long ga = (unsigned long long)(size_t)(src + i * 4);
            const unsigned la = ldsRow + (unsigned)(i * 16);
            asm volatile("global_load_async_to_lds_b128 %0, %1, off"
                         :: "v"(la), "v"(ga) : "memory");
        }
        wait_async_zero();
        __syncthreads();                         // chunk visible to all waves

        for (int kk = 0; kk < 64; kk += 4) {
            const int row = kk + half * 2;
            const size_t wofs = (size_t)(kc * 64 + row) * kV + n0;
            v2f bv;
            bv.x = out_W[wofs];                  // B[K=row  ][n0]
            bv.y = out_W[wofs + kV];             // B[K=row+1][n0]
            #pragma unroll
            for (int mm = 0; mm < 8; ++mm) {
                const int aidx = (mm * 16 + nl) * LDW + row;
                v2f av;                          // ds_load_b64, bank-clean
                av.x = catL[aidx];               // A[M][K=row]
                av.y = catL[aidx + 1];           // A[M][K=row+1]
                acc[mm] = __builtin_amdgcn_wmma_f32_16x16x4_f32(
                    false, av, false, bv, (short)0, acc[mm], false, false);
            }
        }
    }

    const float ob = out_b[n0];
    #pragma unroll
    for (int mm = 0; mm < 8; ++mm) {
        #pragma unroll
        for (int r = 0; r < 8; ++r) {
            const int brow = mm * 16 + r + half * 8;   // D layout: VGPR r -> M
            logits[(size_t)brow * kV + n0] = acc[mm][r] + ob;
        }
    }
}

// ---------------------------------------------------------------------------
// K6: in-place log_softmax over V per batch row
// ---------------------------------------------------------------------------
__global__ void k6_logsoftmax(float* __restrict__ logits)
{
    const int b = blockIdx.x;
    const int t = threadIdx.x;
    __shared__ float red[256];
    float* row = logits + (size_t)b * kV;

    float m = -INFINITY;
    for (int i = t; i < kV; i += 256) m = fmaxf(m, row[i]);
    red[t] = m;
    __syncthreads();
    for (int off = 128; off > 0; off >>= 1) {
        if (t < off) red[t] = fmaxf(red[t], red[t + off]);
        __syncthreads();
    }
    m = red[0];
    __syncthreads();

    float s = 0.f;
    for (int i = t; i < kV; i += 256) s += __expf(row[i] - m);
    red[t] = s;
    __syncthreads();
    for (int off = 128; off > 0; off >>= 1) {
        if (t < off) red[t] += red[t + off];
        __syncthreads();
    }
    const float lse = m + logf(red[0]);
    for (int i = t; i < kV; i += 256) row[i] = row[i] - lse;
}

// ---------------------------------------------------------------------------
extern "C" void kernel_launch(void* const* d_in, const int* in_sizes, int n_in,
                              void* d_out, int out_size, void* d_ws, size_t ws_size,
                              hipStream_t stream)
{
    const int*   ids    = (const int*)  d_in[0];
    const float* hidden = (const float*)d_in[1];
    const float* enc    = (const float*)d_in[2];
    const float* emb    = (const float*)d_in[3];
    const float* attn_W = (const float*)d_in[4];
    const float* attn_b = (const float*)d_in[5];
    const float* W_ih   = (const float*)d_in[6];
    const float* W_hh   = (const float*)d_in[7];
    const float* b_ih   = (const float*)d_in[8];
    const float* b_hh   = (const float*)d_in[9];
    const float* out_W  = (const float*)d_in[10];
    const float* out_b  = (const float*)d_in[11];

    float* out      = (float*)d_out;
    float* logits   = out;                                     // (B, V)
    float* h_new    = out + (size_t)kB * kV;                   // (1, B, H)
    float* attn_out = out + (size_t)kB * kV + (size_t)kB * kH; // (B, 1, S)

    float* ws       = (float*)d_ws;
    float* u        = ws;                                  // B*H
    float* bias     = u + (size_t)kB * kH;                 // B
    float* scores   = bias + kB;                           // B*S
    float* ctx      = scores + (size_t)kB * kS;            // B*H
    float* splitM   = ctx + (size_t)kB * kH;               // B*8
    float* splitD   = splitM + (size_t)kB * 8;             // B*8
    float* splitCtx = splitD + (size_t)kB * 8;             // B*8*H

    k1_attn_pre<<<kB, 256, 0, stream>>>(hidden, attn_W, attn_b, u, bias);
    k2_attn_fused<<<dim3(kB, 8), 256, 0, stream>>>(enc, u, bias, scores,
                                                   splitM, splitD, splitCtx);
    k3_combine<<<kB, 256, 0, stream>>>(scores, splitM, splitD, splitCtx,
                                       attn_out, ctx);
    k4_gru<<<kB, 256, 0, stream>>>(ids, hidden, emb, ctx, W_ih, W_hh, b_ih, b_hh, h_new);
    k5_logits_wmma<<<kV / 128, 256, 0, stream>>>(h_new, ctx, out_W, out_b, logits);
    k6_logsoftmax<<<kB, 256, 0, stream>>>(logits);
}